// PrototypeRepresentationLayer_78615081386036
// MI455X (gfx1250) — compile-verified
//
#include <hip/hip_runtime.h>

#define GAMMA_F   0.1f
#define BIG_F     1e8f
#define T_LEN     128
#define S_LEN     128
#define C_DIM     20
#define D_STRIDE  132   // 128 cols + pad: diag stride 131 == 3 (mod 64) -> conflict-free wavefront
#define NDIAG     (T_LEN + S_LEN - 1)   // 255 == 3 * 85
#define BUF_LEN   130   // guard slot [0] + rows [1..128] + spare

typedef float v2f __attribute__((ext_vector_type(2)));
typedef float v8f __attribute__((ext_vector_type(8)));

// -gamma * logsumexp(-{a,b,c}/gamma), numerically stable, in base-2 hardware units:
//   u = x * (-log2(e)/gamma);  softmin = -gamma*ln2 * (max(u) + log2(sum 2^(u-max)))
__device__ __forceinline__ float softmin3(float a, float b, float c) {
    const float kin  = -1.4426950408889634f / GAMMA_F;   // -log2(e)/gamma
    const float kout = -0.6931471805599453f * GAMMA_F;   // -ln(2)*gamma
    float xa = a * kin;
    float xb = b * kin;
    float xc = c * kin;
    float m = fmaxf(xa, fmaxf(xb, xc));
    float s = __builtin_amdgcn_exp2f(xa - m) +
              __builtin_amdgcn_exp2f(xb - m) +
              __builtin_amdgcn_exp2f(xc - m);
    return kout * (m + __builtin_amdgcn_logf(s));        // v_log_f32 is log2
}

__device__ __forceinline__ void dp_step(const float* __restrict__ bm2,
                                        const float* __restrict__ bm1,
                                        float* __restrict__ bcur,
                                        int d, int i,
                                        const float* __restrict__ myD,
                                        float& result) {
    const int j = d - i;
    if (j >= 0 && j < S_LEN) {
        const float up   = bm1[i];                    // R[i-1][j]   (guard BIG if i==0)
        const float left = bm1[i + 1];                // R[i][j-1]   (init BIG if j==0)
        const float dg   = (d == 0) ? 0.0f : bm2[i];  // R[i-1][j-1] (R[-1][-1]=0)
        const float r = myD[d] + softmin3(up, left, dg);
        bcur[i + 1] = r;
        result = r;
    }
    __syncthreads();
}

__global__ __launch_bounds__(128)
void softdtw_proto_kernel(const float* __restrict__ x,
                          const float* __restrict__ protos,
                          float* __restrict__ out,
                          int P) {
    extern __shared__ float smem[];
    float* sx2  = smem;                       // 128
    float* sy2  = sx2 + T_LEN;                // 128
    float* sD   = sy2 + S_LEN;                // 128*132 = 16896
    float* sbuf = sD  + T_LEN * D_STRIDE;     // 3*130   = 390

    const int bp  = blockIdx.x;
    const int b   = bp / P;
    const int p   = bp - b * P;
    const int tid = threadIdx.x;

    const float* gx = x      + (size_t)b * (T_LEN * C_DIM);
    const float* gy = protos + (size_t)p * (S_LEN * C_DIM);

    // ---- Row squared norms (straight from global; 10KB/matrix, L1-resident) ----
    {
        float ax = 0.0f, ay = 0.0f;
        const float* rx = gx + tid * C_DIM;
        const float* ry = gy + tid * C_DIM;
#pragma unroll
        for (int c = 0; c < C_DIM; ++c) {
            ax = fmaf(rx[c], rx[c], ax);
            ay = fmaf(ry[c], ry[c], ay);
        }
        sx2[tid] = ax;
        sy2[tid] = ay;
    }
    // DP diagonal buffers start as BIG (never-written slots must stay BIG)
    for (int idx = tid; idx < 3 * BUF_LEN; idx += 128) sbuf[idx] = BIG_F;
    __syncthreads();

    // ---- Phase 1: D[t][s] = |x_t|^2 + |y_s|^2 - 2 <x_t, y_s> via f32 WMMA 16x16x4 ----
    // 64 tiles of 16x16; 4 waves x 16 tiles; K=20 -> 5 chained WMMAs per tile.
    {
        const int wave = tid >> 5;
        const int lane = tid & 31;
        const int mrow = lane & 15;   // row-in-tile for A/B fragments
        const int hi   = lane >> 4;   // K half-select ({0,1} vs {2,3} within K-group)

        for (int tt = wave; tt < 64; tt += 4) {
            const int ti = tt >> 3;
            const int tj = tt & 7;
            v8f acc = {0.f, 0.f, 0.f, 0.f, 0.f, 0.f, 0.f, 0.f};
#pragma unroll
            for (int kk = 0; kk < C_DIM; kk += 4) {
                const float* pa = gx + (ti * 16 + mrow) * C_DIM + kk + 2 * hi;
                const float* pb = gy + (tj * 16 + mrow) * C_DIM + kk + 2 * hi;
                v2f a;  a.x  = pa[0]; a.y  = pa[1];
                v2f bb; bb.x = pb[0]; bb.y = pb[1];
                acc = __builtin_amdgcn_wmma_f32_16x16x4_f32(
                    false, a, false, bb, (short)0, acc, false, false);
            }
            // C/D layout: VGPR v, lanes 0-15 -> M=v; lanes 16-31 -> M=v+8; N = lane&15
#pragma unroll
            for (int v = 0; v < 8; ++v) {
                const int row = ti * 16 + v + 8 * hi;   // t index
                const int col = tj * 16 + mrow;         // s index
                sD[row * D_STRIDE + col] = sx2[row] + sy2[col] - 2.0f * acc[v];
            }
        }
    }
    __syncthreads();

    // ---- Phase 2: anti-diagonal wavefront DP, triple-buffered diagonals ----
    float* b0 = sbuf;
    float* b1 = sbuf + BUF_LEN;
    float* b2 = sbuf + 2 * BUF_LEN;

    const int i = tid;                               // this thread owns row i
    const float* myD = sD + i * (D_STRIDE - 1);      // myD[d] == sD[i*D_STRIDE + (d-i)]
    float result = 0.0f;

    for (int d = 0; d < NDIAG; d += 3) {             // 255 = 3*85, rotation period 3
        dp_step(b0, b1, b2, d,     i, myD, result);
        dp_step(b1, b2, b0, d + 1, i, myD, result);
        dp_step(b2, b0, b1, d + 2, i, myD, result);
    }

    if (i == T_LEN - 1) out[bp] = result;            // R[T-1][S-1]
}

extern "C" void kernel_launch(void* const* d_in, const int* in_sizes, int n_in,
                              void* d_out, int out_size, void* d_ws, size_t ws_size,
                              hipStream_t stream) {
    const float* x      = (const float*)d_in[0];   // (B, T, C) f32
    const float* protos = (const float*)d_in[1];   // (P, S, C) f32
    float* out = (float*)d_out;                    // (B, P) f32

    const int B = in_sizes[0] / (T_LEN * C_DIM);   // 64
    const int P = in_sizes[1] / (S_LEN * C_DIM);   // 32

    const size_t smem_bytes =
        (size_t)(T_LEN + S_LEN + T_LEN * D_STRIDE + 3 * BUF_LEN) * sizeof(float); // 70,168 B -> 4 WGs/WGP

    (void)hipFuncSetAttribute((const void*)softdtw_proto_kernel,
                              hipFuncAttributeMaxDynamicSharedMemorySize,
                              (int)smem_bytes);

    softdtw_proto_kernel<<<B * P, 128, smem_bytes, stream>>>(x, protos, out, P);
}